// HolographicQKV_86294482911688
// MI455X (gfx1250) — compile-verified
//
#include <hip/hip_runtime.h>
#include <hip/hip_bf16.h>

#define D_DIM 1024
#define S_DIM 4096
#define B_DIM 4

typedef __attribute__((ext_vector_type(2))) float v2f;
typedef __attribute__((ext_vector_type(8))) float v8f;

#if defined(__AMDGCN__) && __has_builtin(__builtin_amdgcn_global_load_async_to_lds_b128)
#define HAVE_ASYNC_LDS 1
#else
#define HAVE_ASYNC_LDS 0
#endif

#if HAVE_ASYNC_LDS
// Builtin signature (from hipcc diagnostic): pointers to int vector_size(16),
// global (AS1) source first, LDS (AS3) destination second, imm offset, imm cpol.
typedef int v4i_vs __attribute__((vector_size(4 * sizeof(int))));
typedef __attribute__((address_space(1))) v4i_vs* g_v4i_p;
typedef __attribute__((address_space(3))) v4i_vs* l_v4i_p;
#endif

// 16-byte global -> LDS copy. Async (ASYNCcnt-tracked, no VGPR roundtrip) when
// the toolchain exposes the gfx1250 builtin, else plain load + ds_store_b128.
__device__ __forceinline__ void cp16(const float* __restrict__ g, float* l)
{
#if HAVE_ASYNC_LDS
    __builtin_amdgcn_global_load_async_to_lds_b128(
        (g_v4i_p)g, (l_v4i_p)l, 0, 0);
#else
    *(float4*)l = *(const float4*)g;
#endif
}

__device__ __forceinline__ void async_wait_all()
{
#if HAVE_ASYNC_LDS
    asm volatile("s_wait_asynccnt 0x0" ::: "memory");
#endif
}

// ---------------------------------------------------------------------------
// f32 WMMA GEMM with LDS double-buffered async staging + split-K over grid.z:
//   Dst[z][M,N] = op(A)[:, kz:kz+kPerSlice] * B[kz:kz+kPerSlice, :]
//     (kz = z*kPerSlice; Dst advances by M*N per z; z=1 slice => plain GEMM)
//   TA=false: op(A)[m][k] = A[m*lda + k]     TA=true: op(A)[m][k] = A[k*lda + m]
// Block = 256 threads (8 waves). Block tile 128x128, wave tile 32x64 (2x4 WMMA).
// M,N multiples of 128; kPerSlice multiple of 16 (K-chunk).
// ---------------------------------------------------------------------------
#define KC 16
#define LDB_S 136   // K-major panel stride (half-waves hit disjoint bank ranges)
#define LDA_S 20    // M-major A panel stride (lane start banks distinct mod 64)

template <bool TA>
__global__ __launch_bounds__(256)
void gemm_wmma_f32(const float* __restrict__ A, const float* __restrict__ B,
                   float* __restrict__ Dst, int M, int N, int kPerSlice,
                   int lda, int ldb, int ldd,
                   const float* __restrict__ rowAdd)
{
    __shared__ float sA[2][128 * LDA_S];  // NN: [m][k] ; TA reuses as [k][LDB_S]
    __shared__ float sB[2][KC * LDB_S];

    const int tid  = threadIdx.x;
    const int wave = tid >> 5;
    const int lane = tid & 31;
    const int half = lane >> 4;
    const int l16  = lane & 15;

    const int bm0 = blockIdx.y * 128;
    const int bn0 = blockIdx.x * 128;
    const int wr  = (wave >> 1) * 32;   // wave tile row in block
    const int wc  = (wave & 1) * 64;    // wave tile col in block

    const int kStart = blockIdx.z * kPerSlice;
    Dst += (size_t)blockIdx.z * M * N;

    v8f acc[2][4];
#pragma unroll
    for (int i = 0; i < 2; ++i)
#pragma unroll
        for (int j = 0; j < 4; ++j)
            acc[i][j] = (v8f){0.f, 0.f, 0.f, 0.f, 0.f, 0.f, 0.f, 0.f};

    // ---- panel staging for chunk starting at kc into buffer `buf` ----------
    auto stage = [&](int buf, int kc) {
        // B panel: KC x 128, K-major, contiguous 16B segments
#pragma unroll
        for (int it = 0; it < 2; ++it) {
            const int seg  = tid + 256 * it;     // 512 segments = 16 x 32
            const int row  = seg >> 5;
            const int col4 = (seg & 31) * 4;
            cp16(B + (size_t)(kc + row) * ldb + bn0 + col4,
                 &sB[buf][row * LDB_S + col4]);
        }
        if (TA) {
            // A panel: KC x 128, K-major (global already K-major)
#pragma unroll
            for (int it = 0; it < 2; ++it) {
                const int seg  = tid + 256 * it;
                const int row  = seg >> 5;
                const int col4 = (seg & 31) * 4;
                cp16(A + (size_t)(kc + row) * lda + bm0 + col4,
                     &sA[buf][row * LDB_S + col4]);
            }
        } else {
            // A panel: 128 x KC, M-major (global already M-major)
            const int m    = tid >> 1;
            const int koff = (tid & 1) * 8;
#pragma unroll
            for (int q = 0; q < 2; ++q) {
                const int off = koff + 4 * q;
                cp16(A + (size_t)(bm0 + m) * lda + kc + off,
                     &sA[buf][m * LDA_S + off]);
            }
        }
    };

    const int nChunks = kPerSlice / KC;

    stage(0, kStart);
    async_wait_all();
    __syncthreads();

    int cur = 0;
    for (int c = 0; c < nChunks; ++c) {
        const int next = cur ^ 1;
        if (c + 1 < nChunks) stage(next, kStart + (c + 1) * KC);  // prefetch

        // ---- compute this chunk from LDS ----
#pragma unroll
        for (int ks = 0; ks < KC; ks += 4) {
            const int kk = ks + 2 * half;   // this half-wave's K pair

            v2f af[2];
#pragma unroll
            for (int i = 0; i < 2; ++i) {
                const int m = wr + 16 * i + l16;
                if (TA) {
                    af[i].x = sA[cur][(kk + 0) * LDB_S + m];
                    af[i].y = sA[cur][(kk + 1) * LDB_S + m];
                } else {
                    af[i] = *(const v2f*)&sA[cur][m * LDA_S + kk];
                }
            }
            v2f bf[4];
#pragma unroll
            for (int j = 0; j < 4; ++j) {
                const int n = wc + 16 * j + l16;
                bf[j].x = sB[cur][(kk + 0) * LDB_S + n];
                bf[j].y = sB[cur][(kk + 1) * LDB_S + n];
            }
#pragma unroll
            for (int i = 0; i < 2; ++i)
#pragma unroll
                for (int j = 0; j < 4; ++j)
                    acc[i][j] = __builtin_amdgcn_wmma_f32_16x16x4_f32(
                        false, af[i], false, bf[j],
                        (short)0, acc[i][j], false, false);
        }

        async_wait_all();     // prefetch landed
        __syncthreads();      // everyone done reading `cur`, `next` visible
        cur = next;
    }

    // ---- epilogue: C/D layout lane<16: M=v, lane>=16: M=v+8 ; N=l16 --------
#pragma unroll
    for (int i = 0; i < 2; ++i) {
#pragma unroll
        for (int j = 0; j < 4; ++j) {
            const int col = bn0 + wc + 16 * j + l16;
            const float add = rowAdd ? rowAdd[col] : 0.f;
#pragma unroll
            for (int v = 0; v < 8; ++v) {
                const int row = bm0 + wr + 16 * i + v + 8 * half;
                Dst[(size_t)row * ldd + col] = acc[i][j][v] + add;
            }
        }
    }
}

// ---------------------------------------------------------------------------
// Split-K partial reduction (deterministic; no float atomics)
// ---------------------------------------------------------------------------
__global__ void reduceK_kernel(const float* __restrict__ P, float* __restrict__ G,
                               int nslice)
{
    const size_t idx = (size_t)blockIdx.x * blockDim.x + threadIdx.x;
    const size_t MAT = (size_t)D_DIM * D_DIM;
    float s = 0.f;
    for (int z = 0; z < nslice; ++z) s += P[z * MAT + idx];
    G[idx] = s;
}

// ---------------------------------------------------------------------------
// Small helper kernels (O(D) or O(D^2) threads, negligible cost)
// ---------------------------------------------------------------------------
__global__ void colsum_kernel(const float* __restrict__ X, float* __restrict__ xs)
{
    const int d = blockIdx.x * blockDim.x + threadIdx.x;
    float s = 0.f;
    for (int t = 0; t < S_DIM; ++t) s += X[(size_t)t * D_DIM + d];
    xs[d] = s;
}

__global__ void matvecT2_kernel(const float* __restrict__ Wk,
                                const float* __restrict__ Wv,
                                const float* __restrict__ xs,
                                float* __restrict__ u, float* __restrict__ w)
{
    const int m = blockIdx.x * blockDim.x + threadIdx.x;
    float su = 0.f, sw = 0.f;
    for (int k = 0; k < D_DIM; ++k) {
        const float xv = xs[k];
        su += Wk[(size_t)k * D_DIM + m] * xv;
        sw += Wv[(size_t)k * D_DIM + m] * xv;
    }
    u[m] = su;
    w[m] = sw;
}

// C += u*bv^T + bk*w^T + S*bk*bv^T   (exact bias terms of K^T V)
__global__ void rank1_kernel(float* __restrict__ C,
                             const float* __restrict__ u,
                             const float* __restrict__ bk,
                             const float* __restrict__ w,
                             const float* __restrict__ bv)
{
    const int idx = blockIdx.x * blockDim.x + threadIdx.x;
    const int j = idx >> 10;
    const int m = idx & (D_DIM - 1);
    C[idx] += u[j] * bv[m] + bk[j] * (w[m] + (float)S_DIM * bv[m]);
}

// s[d] = sum_j C[j][(d-j) mod D]
__global__ void sred_kernel(const float* __restrict__ C, float* __restrict__ s)
{
    const int d = blockIdx.x * blockDim.x + threadIdx.x;
    float acc = 0.f;
    for (int j = 0; j < D_DIM; ++j)
        acc += C[(size_t)j * D_DIM + ((d - j) & (D_DIM - 1))];
    s[d] = acc;
}

// Am[j][d] = s[(j+d) mod D]
__global__ void buildA_kernel(const float* __restrict__ s, float* __restrict__ Am)
{
    const int idx = blockIdx.x * blockDim.x + threadIdx.x;
    const int j = idx >> 10;
    const int d = idx & (D_DIM - 1);
    Am[idx] = s[(j + d) & (D_DIM - 1)];
}

// r[d] = sum_j bq[j]*s[(j+d) mod D]
__global__ void rvec_kernel(const float* __restrict__ bq,
                            const float* __restrict__ s, float* __restrict__ r)
{
    const int d = blockIdx.x * blockDim.x + threadIdx.x;
    float acc = 0.f;
    for (int j = 0; j < D_DIM; ++j)
        acc += bq[j] * s[(j + d) & (D_DIM - 1)];
    r[d] = acc;
}

// ---------------------------------------------------------------------------
// Per batch b:
//   G = Xb^T Xb  (split-K=8) ; T = Wk^T G ; C = T Wv (+bias rank-1)
//   s = anti-diag reduce C ; Am = circulant(s) ; Wp = Wq Am
//   Out_b = Xb Wp + 1*r^T
// ---------------------------------------------------------------------------
extern "C" void kernel_launch(void* const* d_in, const int* in_sizes, int n_in,
                              void* d_out, int out_size, void* d_ws, size_t ws_size,
                              hipStream_t stream)
{
    const float* x  = (const float*)d_in[0];
    const float* Wq = (const float*)d_in[1];
    const float* bq = (const float*)d_in[2];
    const float* Wk = (const float*)d_in[3];
    const float* bk = (const float*)d_in[4];
    const float* Wv = (const float*)d_in[5];
    const float* bv = (const float*)d_in[6];
    float* out = (float*)d_out;

    float* wsf = (float*)d_ws;
    const size_t MAT = (size_t)D_DIM * D_DIM;
    const int NSLICE = 8;
    float* Gp = wsf;                    // split-K partials: 8 * 4 MB
    float* G  = wsf + (size_t)NSLICE * MAT;
    float* T  = G + MAT;
    float* Cm = T + MAT;
    float* Am = Cm + MAT;
    float* Wp = Am + MAT;
    float* xs = Wp + MAT;
    float* u  = xs + D_DIM;
    float* w  = u + D_DIM;
    float* sv = w + D_DIM;
    float* r  = sv + D_DIM;

    const dim3 blk(256);
    const dim3 gridDD(D_DIM / 128, D_DIM / 128, 1);        // 8 x 8
    const dim3 gridGram(D_DIM / 128, D_DIM / 128, NSLICE); // 8 x 8 x 8
    const dim3 gridSD(D_DIM / 128, S_DIM / 128, 1);        // 8 x 32

    for (int b = 0; b < B_DIM; ++b) {
        const float* Xb = x + (size_t)b * S_DIM * D_DIM;
        float*       Ob = out + (size_t)b * S_DIM * D_DIM;

        // G = Xb^T Xb (M=N=1024, K=4096 split 8 ways), then reduce
        gemm_wmma_f32<true><<<gridGram, blk, 0, stream>>>(
            Xb, Xb, Gp, D_DIM, D_DIM, S_DIM / NSLICE, D_DIM, D_DIM, D_DIM, nullptr);
        reduceK_kernel<<<(int)(MAT / 256), 256, 0, stream>>>(Gp, G, NSLICE);

        colsum_kernel<<<D_DIM / 256, 256, 0, stream>>>(Xb, xs);
        matvecT2_kernel<<<D_DIM / 256, 256, 0, stream>>>(Wk, Wv, xs, u, w);

        // T = Wk^T G ; C = T Wv
        gemm_wmma_f32<true><<<gridDD, blk, 0, stream>>>(
            Wk, G, T, D_DIM, D_DIM, D_DIM, D_DIM, D_DIM, D_DIM, nullptr);
        gemm_wmma_f32<false><<<gridDD, blk, 0, stream>>>(
            T, Wv, Cm, D_DIM, D_DIM, D_DIM, D_DIM, D_DIM, D_DIM, nullptr);

        rank1_kernel<<<(D_DIM * D_DIM) / 256, 256, 0, stream>>>(Cm, u, bk, w, bv);
        sred_kernel<<<D_DIM / 256, 256, 0, stream>>>(Cm, sv);
        buildA_kernel<<<(D_DIM * D_DIM) / 256, 256, 0, stream>>>(sv, Am);
        rvec_kernel<<<D_DIM / 256, 256, 0, stream>>>(bq, sv, r);

        // Wp = Wq Am ; Out_b = Xb Wp + 1*r^T
        gemm_wmma_f32<false><<<gridDD, blk, 0, stream>>>(
            Wq, Am, Wp, D_DIM, D_DIM, D_DIM, D_DIM, D_DIM, D_DIM, nullptr);
        gemm_wmma_f32<false><<<gridSD, blk, 0, stream>>>(
            Xb, Wp, Ob, S_DIM, D_DIM, D_DIM, D_DIM, D_DIM, D_DIM, r);
    }
}